// SpatialCorrelationLossModule_48919677501788
// MI455X (gfx1250) — compile-verified
//
#include <hip/hip_runtime.h>
#include <stdint.h>
#include <math.h>

typedef __attribute__((ext_vector_type(2))) float v2f;
typedef __attribute__((ext_vector_type(8))) float v8f;

#define NRUNS 10
#define KU    500      // units per neighborhood
#define KPAD  512
#define BF    128      // feature count
#define UTOT  100000   // total units
#define LDSW  68       // 64 + 4 pad -> conflict-free (68 % 64 == 4)

// ---------------- workspace layout (bytes from base) ----------------
// 0     : int    choice[16]
// 64    : double sums[10][5]          (zeroed each launch)
// 512   : float  inv_s[10*512]
// 20992 : float  px   [10*512]
// 41472 : float  py   [10*512]
// 61952 : float  xc   [10*512*128]
// total ~2.68 MB

// ------------------------- threefry2x32 -------------------------
__device__ inline void tf_block(uint32_t k0, uint32_t k1, uint32_t x0, uint32_t x1,
                                uint32_t* o0, uint32_t* o1) {
    const uint32_t ks2 = k0 ^ k1 ^ 0x1BD11BDAu;
    uint32_t v0 = x0 + k0, v1 = x1 + k1;
#define TFR(r) { v0 += v1; v1 = (v1 << (r)) | (v1 >> (32 - (r))); v1 ^= v0; }
    TFR(13) TFR(15) TFR(26) TFR(6)   v0 += k1;  v1 += ks2 + 1u;
    TFR(17) TFR(29) TFR(16) TFR(24)  v0 += ks2; v1 += k0 + 2u;
    TFR(13) TFR(15) TFR(26) TFR(6)   v0 += k0;  v1 += k1 + 3u;
    TFR(17) TFR(29) TFR(16) TFR(24)  v0 += k1;  v1 += ks2 + 4u;
    TFR(13) TFR(15) TFR(26) TFR(6)   v0 += ks2; v1 += k0 + 5u;
#undef TFR
    *o0 = v0; *o1 = v1;
}

__global__ void pass_init(int* choice, double* sums) {
    int t = threadIdx.x;
    if (t < NRUNS * 5) sums[t] = 0.0;
    if (t == 0) {
        // split(key=(0,42)) -> k1, k2  (counts [0,1,2,3]; pairs (0,2),(1,3))
        uint32_t a0, a1, b0, b1, o0, o1;
        tf_block(0u, 42u, 0u, 2u, &a0, &b0);
        tf_block(0u, 42u, 1u, 3u, &a1, &b1);
        // random_bits(k,32,(10,)): counts iota(10) -> pairs (i, i+5)
        uint32_t hi[10], lo[10];
        for (int i = 0; i < 5; i++) {
            tf_block(a0, a1, (uint32_t)i, (uint32_t)(i + 5), &o0, &o1);
            hi[i] = o0; hi[i + 5] = o1;
            tf_block(b0, b1, (uint32_t)i, (uint32_t)(i + 5), &o0, &o1);
            lo[i] = o0; lo[i + 5] = o1;
        }
        // randint span=100: multiplier = ((2^16 % 100)^2) % 100 = 96
        for (int i = 0; i < NRUNS; i++)
            choice[i] = (int)(((hi[i] % 100u) * 96u + (lo[i] % 100u)) % 100u);
    }
}

// ------------- gather + center + inv-norm + positions -------------
__global__ void pass_center(const float* __restrict__ feat,
                            const float* __restrict__ pos,
                            const int*   __restrict__ nbr,
                            const int*   __restrict__ choice,
                            float* __restrict__ inv_s,
                            float* __restrict__ px, float* __restrict__ py,
                            float* __restrict__ xc) {
    int wave = threadIdx.x >> 5;               // 8 waves/block, 1 unit/wave
    int lane = threadIdx.x & 31;
    int run  = blockIdx.x >> 6;                // 64 blocks per run
    int u    = ((blockIdx.x & 63) << 3) + wave; // 0..511
    float* xcr = xc + ((size_t)run * KPAD + u) * BF;
    int base = run * KPAD + u;
    if (u >= KU) {                              // zero padding rows
        for (int j = lane; j < BF; j += 32) xcr[j] = 0.f;
        if (lane == 0) { inv_s[base] = 0.f; px[base] = 0.f; py[base] = 0.f; }
        return;
    }
    int ind = nbr[choice[run] * KU + u];
    float v[4]; float s = 0.f;
#pragma unroll
    for (int j = 0; j < 4; j++) {
        v[j] = feat[(size_t)(lane + 32 * j) * UTOT + ind];
        s += v[j];
    }
    for (int m = 16; m >= 1; m >>= 1) s += __shfl_xor(s, m, 32);
    float mean = s * (1.f / 128.f);
    float sq = 0.f;
#pragma unroll
    for (int j = 0; j < 4; j++) {
        float c = v[j] - mean;
        xcr[lane + 32 * j] = c;
        sq += c * c;
    }
    for (int m = 16; m >= 1; m >>= 1) sq += __shfl_xor(sq, m, 32);
    if (lane == 0) {
        inv_s[base] = (sq > 0.f) ? 1.f / sqrtf(sq) : 0.f;
        px[base] = pos[2 * (size_t)ind];
        py[base] = pos[2 * (size_t)ind + 1];
    }
}

// ------------- WMMA Gram matrix + fused Pearson sums -------------
__global__ void __launch_bounds__(512) pass_gemm(const float* __restrict__ xc,
                                                 const float* __restrict__ inv_s,
                                                 const float* __restrict__ px,
                                                 const float* __restrict__ py,
                                                 double* __restrict__ sums) {
    __shared__ float At[64 * LDSW];
    __shared__ float Bt[64 * LDSW];

    int run = blockIdx.x / 36;
    int t   = blockIdx.x % 36;
    int I = 0;
    while ((I + 1) * (I + 2) / 2 <= t) I++;     // lower-tri macro tile (I>=J)
    int J = t - I * (I + 1) / 2;

    const float* xcr = xc + (size_t)run * KPAD * BF;
    int wv   = threadIdx.x >> 5, lane = threadIdx.x & 31;
    int ti   = (wv >> 2) * 16,   tj   = (wv & 3) * 16;
    int m    = lane & 15;
    int ksel = (lane >> 4) << 1;                // 0 or 2

    v8f acc = {};
    for (int kc = 0; kc < 2; kc++) {            // two K=64 chunks
        // stage 64 rows x 64 cols of each side into LDS (coalesced float4)
        for (int c = 0; c < 2; c++) {
            int linear = c * 512 + threadIdx.x; // 0..1023 float4 slots
            int row = linear >> 4;              // 16 float4 per row
            int col = (linear & 15) << 2;
            float4 a = *(const float4*)(xcr + (size_t)(I * 64 + row) * BF + kc * 64 + col);
            float4 b = *(const float4*)(xcr + (size_t)(J * 64 + row) * BF + kc * 64 + col);
            *(float4*)&At[row * LDSW + col] = a;
            *(float4*)&Bt[row * LDSW + col] = b;
        }
        __syncthreads();
        const float* arow = &At[(ti + m) * LDSW + ksel];
        const float* brow = &Bt[(tj + m) * LDSW + ksel];
#pragma unroll
        for (int k0 = 0; k0 < 64; k0 += 4) {
            v2f a = *(const v2f*)(arow + k0);
            v2f b = *(const v2f*)(brow + k0);
            acc = __builtin_amdgcn_wmma_f32_16x16x4_f32(
                false, a, false, b, (short)0, acc, false, false);
        }
        __syncthreads();
    }

    // fused epilogue: corr + dist_sim + Pearson partial sums
    int i0 = I * 64 + ti, j0 = J * 64 + tj;
    const float* invs = inv_s + run * KPAD;
    const float* pxr  = px + run * KPAD;
    const float* pyr  = py + run * KPAD;
    int j = j0 + (lane & 15);
    int mrow = (lane >> 4) * 8;
    double sa = 0, sb = 0, saa = 0, sbb = 0, sab = 0;
#pragma unroll
    for (int v = 0; v < 8; v++) {
        int i = i0 + mrow + v;
        if (i < KU && j < i) {                  // strict lower triangle only
            float resp = acc[v] * invs[i] * invs[j];
            float dx = pxr[i] - pxr[j], dy = pyr[i] - pyr[j];
            float bb = 1.f / (sqrtf(dx * dx + dy * dy) + 1.f);
            double rd = (double)resp, bd = (double)bb;
            sa += rd; sb += bd;
            saa += rd * rd; sbb += bd * bd; sab += rd * bd;
        }
    }
    for (int msk = 16; msk >= 1; msk >>= 1) {
        sa  += __shfl_xor(sa,  msk, 32);
        sb  += __shfl_xor(sb,  msk, 32);
        saa += __shfl_xor(saa, msk, 32);
        sbb += __shfl_xor(sbb, msk, 32);
        sab += __shfl_xor(sab, msk, 32);
    }
    if (lane == 0) {
        double* s5 = sums + run * 5;
        atomicAdd(&s5[0], sa);
        atomicAdd(&s5[1], sb);
        atomicAdd(&s5[2], saa);
        atomicAdd(&s5[3], sbb);
        atomicAdd(&s5[4], sab);
    }
}

// ------------------------- final reduce -------------------------
__global__ void pass_final(const double* __restrict__ sums, float* __restrict__ out) {
    const double n = 124750.0;                  // 500*499/2
    double acc = 0.0;
    for (int r = 0; r < NRUNS; r++) {
        const double* s = sums + r * 5;
        double sa = s[0], sb = s[1], saa = s[2], sbb = s[3], sab = s[4];
        double cab = sab - sa * sb / n;
        double ca  = saa - sa * sa / n;
        double cb  = sbb - sb * sb / n;
        double rr  = cab / sqrt(ca * cb);
        acc += (1.0 - rr) * 0.5;
    }
    out[0] = (float)(acc / NRUNS);
}

extern "C" void kernel_launch(void* const* d_in, const int* in_sizes, int n_in,
                              void* d_out, int out_size, void* d_ws, size_t ws_size,
                              hipStream_t stream) {
    const float* feat = (const float*)d_in[0];  // (128, 100000) f32
    const float* pos  = (const float*)d_in[1];  // (100000, 2)  f32
    const int*   nbr  = (const int*)d_in[2];    // (100, 500)   i32

    char*   ws     = (char*)d_ws;
    int*    choice = (int*)ws;                       // 64 B
    double* sums   = (double*)(ws + 64);             // 400 B
    float*  inv_s  = (float*)(ws + 512);             // 20480 B
    float*  px     = (float*)(ws + 512 + 20480);
    float*  py     = (float*)(ws + 512 + 40960);
    float*  xc     = (float*)(ws + 512 + 61440);     // 10*512*128 f32

    pass_init  <<<1,   64,  0, stream>>>(choice, sums);
    pass_center<<<640, 256, 0, stream>>>(feat, pos, nbr, choice, inv_s, px, py, xc);
    pass_gemm  <<<360, 512, 0, stream>>>(xc, inv_s, px, py, sums);
    pass_final <<<1,   1,   0, stream>>>(sums, (float*)d_out);
}